// CMFT_41532333752881
// MI455X (gfx1250) — compile-verified
//
#include <hip/hip_runtime.h>

// exp-weighted attention contraction:
//   X[n,c,i] = sum_j exp(a[n,c,i] * b[n,c,j]) * v[n,c,j],  HW = 1024, N*C = 32
//
// Fused exp-outer-product GEMV via V_WMMA_F32_16X16X4_F32 (f32 mandatory:
// exp(a*b) reaches ~1e11, f16 overflows). B operand broadcasts v[j..j+3]
// across all 16 columns so every accumulator column holds the row sums.
//
// VALU minimization: hardware exp is exp2, so stage b pre-scaled by log2(e)
// into LDS and use native exp2 -> hot loop is 1 v_pk_mul + 2 v_exp_f32 per
// K=4 WMMA step per lane (transcendental-bound, matrix pipe co-executes).

typedef float v2f __attribute__((ext_vector_type(2)));
typedef float v8f __attribute__((ext_vector_type(8)));

#define HW 1024
#define ROWS_PER_BLOCK 128   // 8 waves * 16-row tile per wave
#define THREADS 256
#define LOG2E 1.4426950408889634f

__global__ __launch_bounds__(THREADS) void expattn_wmma_f32_kernel(
    const float* __restrict__ fxA,   // a : query-side scalar per (nc,i)
    const float* __restrict__ fyA,   // b : key-side vector per (nc,j)
    const float* __restrict__ fyB,   // v : value vector per (nc,j)
    float* __restrict__ out)
{
    __shared__ __align__(16) float b_s[HW];   // holds b * log2e
    __shared__ __align__(16) float v_s[HW];

    const int bid   = blockIdx.x;
    const int nc    = bid >> 3;        // which (n,c) slice (0..31)
    const int chunk = bid & 7;         // which 128-row chunk of i (0..7)
    const int base  = nc * HW;
    const int tid   = threadIdx.x;

    // Stage b (pre-scaled by log2e) and v into LDS: 256 threads x float4.
    {
        float4 b4 = ((const float4*)(fyA + base))[tid];
        b4.x *= LOG2E; b4.y *= LOG2E; b4.z *= LOG2E; b4.w *= LOG2E;
        ((float4*)b_s)[tid] = b4;
        ((float4*)v_s)[tid] = ((const float4*)(fyB + base))[tid];
    }
    __syncthreads();

    const int wave = tid >> 5;
    const int lane = tid & 31;
    const int half = lane >> 4;        // 0: lanes 0-15 (K=0,1), 1: lanes 16-31 (K=2,3)
    const int m    = lane & 15;        // row within 16x16 tile
    const int row  = chunk * ROWS_PER_BLOCK + wave * 16 + m;

    const float a_row = fxA[base + row];
    const int   kk    = half << 1;     // k offset this half-wave covers: 0 or 2

    v8f acc = {};

#pragma unroll 8
    for (int j = 0; j < HW; j += 4) {
        v2f bp = *(const v2f*)(b_s + j + kk);   // log2e*b[j+kk], log2e*b[j+kk+1]
        v2f vp = *(const v2f*)(v_s + j + kk);   // v[j+kk], v[j+kk+1]

        v2f aFrag;
        aFrag.x = __builtin_amdgcn_exp2f(a_row * bp.x);  // exp(a*b) via native exp2
        aFrag.y = __builtin_amdgcn_exp2f(a_row * bp.y);

        // D = A(16x4 f32 exp tile) * B(4x16 f32, v broadcast over columns) + C
        acc = __builtin_amdgcn_wmma_f32_16x16x4_f32(
            false, aFrag, false, vp, (short)0, acc, false, false);
    }

    // C/D layout: lane l<16 vgpr r = D[r][l]; lane l+16 vgpr r = D[8+r][l].
    // All columns identical -> lane 0 has rows 0-7, lane 16 rows 8-15 (col 0).
    if (m == 0) {
        float* o = out + base + chunk * ROWS_PER_BLOCK + wave * 16 + half * 8;
        *(float4*)(o)     = make_float4(acc[0], acc[1], acc[2], acc[3]);
        *(float4*)(o + 4) = make_float4(acc[4], acc[5], acc[6], acc[7]);
    }
}

extern "C" void kernel_launch(void* const* d_in, const int* in_sizes, int n_in,
                              void* d_out, int out_size, void* d_ws, size_t ws_size,
                              hipStream_t stream) {
    const float* fxA = (const float*)d_in[0];
    const float* fyA = (const float*)d_in[1];
    const float* fyB = (const float*)d_in[2];
    float* out = (float*)d_out;

    // out_size = N*C*HW = 32768 ; each block covers 128 output rows.
    const int blocks = out_size / ROWS_PER_BLOCK;   // 256 blocks, 8 waves each
    expattn_wmma_f32_kernel<<<blocks, THREADS, 0, stream>>>(fxA, fyA, fyB, out);
}